// EntropyLoss_14061722927696
// MI455X (gfx1250) — compile-verified
//
#include <hip/hip_runtime.h>
#include <math.h>

// ---------------------------------------------------------------------------
// EntropyLoss: out = mean_i( -sum_j p_ij * log(p_ij + 1e-8) ),  p = x / colnorm
// Fused single-pass formulation (saves 256MB of HBM traffic vs 2-pass):
//   S  = sum_j inv_j * ( colsum(x*logx)_j + log(inv_j) * colsum(x)_j )
//   out = -(S + N*1e-8) / R      (analytic eps-in-log correction, err ~1e-7 rel)
// Pass 1 streams 256MB once via gfx1250 async global->LDS (ASYNCcnt) pipeline.
// ---------------------------------------------------------------------------

#ifndef __has_builtin
#define __has_builtin(x) 0
#endif

#if __has_builtin(__builtin_amdgcn_global_load_async_to_lds_b128)
#define HAS_ASYNC_LDS 1
#else
#define HAS_ASYNC_LDS 0
#endif

#if __has_builtin(__builtin_amdgcn_s_wait_asynccnt)
#define WAIT_ASYNC(n) __builtin_amdgcn_s_wait_asynccnt(n)
#else
#define WAIT_ASYNC(n) asm volatile("s_wait_asynccnt %0" ::"n"(n) : "memory")
#endif

#if __has_builtin(__builtin_amdgcn_s_wait_dscnt)
#define WAIT_DS(n) __builtin_amdgcn_s_wait_dscnt(n)
#else
#define WAIT_DS(n) asm volatile("s_wait_dscnt %0" ::"n"(n) : "memory")
#endif

#define CFENCE() asm volatile("" ::: "memory")

#define COLS 1024
#define ROWS_PER_CHUNK 512
#define ITERS 128 /* rows per thread = ROWS_PER_CHUNK / 4 subgroups */
#define DEPTH 4
#define NPH (ITERS / DEPTH)

typedef int v4i_t __attribute__((ext_vector_type(4)));
typedef __attribute__((address_space(1))) v4i_t* gas_v4i_ptr; // prints as __device__
typedef __attribute__((address_space(3))) v4i_t* las_v4i_ptr; // prints as __shared__

#if HAS_ASYNC_LDS
__device__ __forceinline__ void async_ld16(const float* gp, float4* lp) {
  __builtin_amdgcn_global_load_async_to_lds_b128((gas_v4i_ptr)gp,
                                                 (las_v4i_ptr)lp, 0, 0);
}
#endif

__device__ __forceinline__ void acc1(float v, float& a1, float& a2, float& a3) {
  a1 += v;
  a2 = fmaf(v, v, a2);
  const float lg = (v > 0.0f) ? __logf(v) : 0.0f; // x*log(x) -> 0 as x -> 0
  a3 = fmaf(v, lg, a3);
}

// Grid: (4 column-groups of 256 cols, nchunk row-chunks of 512 rows), 256 thr.
// Thread t: c = t&63 -> 4 consecutive cols (float4), g = t>>6 -> row subgroup.
// Writes per-chunk column partials: partial[chunk][3][1024] (no atomics).
__global__ __launch_bounds__(256) void colstats_kernel(
    const float* __restrict__ x, float* __restrict__ partial) {
  const int t = threadIdx.x;
  const int c = t & 63;
  const int g = t >> 6;
  const int colbase = (int)blockIdx.x * 256 + c * 4;
  const size_t rowbase = (size_t)blockIdx.y * ROWS_PER_CHUNK + (size_t)g;

  float4 s1 = make_float4(0.f, 0.f, 0.f, 0.f);
  float4 s2 = s1, s3 = s1;

  const float* base = x + rowbase * COLS + (size_t)colbase;
  // iteration i touches row rowbase + 4*i  ->  address stride 4*COLS floats

#if HAS_ASYNC_LDS
  __shared__ float4 abuf[2][DEPTH][256];
  int cur = 0;
#pragma unroll
  for (int d = 0; d < DEPTH; ++d)
    async_ld16(base + (size_t)d * (4 * COLS), &abuf[0][d][t]);
  for (int ph = 0; ph < NPH; ++ph) {
    if (ph + 1 < NPH) {
      const float* nb = base + (size_t)(ph + 1) * DEPTH * (4 * COLS);
#pragma unroll
      for (int d = 0; d < DEPTH; ++d)
        async_ld16(nb + (size_t)d * (4 * COLS), &abuf[cur ^ 1][d][t]);
      CFENCE();
      WAIT_ASYNC(DEPTH); // current buffer's DEPTH loads have landed
    } else {
      CFENCE();
      WAIT_ASYNC(0);
    }
    CFENCE();
#pragma unroll
    for (int d = 0; d < DEPTH; ++d) {
      float4 v = abuf[cur][d][t]; // ds_load_b128
      acc1(v.x, s1.x, s2.x, s3.x);
      acc1(v.y, s1.y, s2.y, s3.y);
      acc1(v.z, s1.z, s2.z, s3.z);
      acc1(v.w, s1.w, s2.w, s3.w);
    }
    CFENCE();
    WAIT_DS(0); // LDS reads done before this buffer is overwritten next iter
    CFENCE();
    cur ^= 1;
  }
#else
  for (int i = 0; i < ITERS; i += 8) {
    float4 v[8];
#pragma unroll
    for (int k = 0; k < 8; ++k)
      v[k] = *(const float4*)(base + (size_t)(i + k) * (4 * COLS));
#pragma unroll
    for (int k = 0; k < 8; ++k) {
      acc1(v[k].x, s1.x, s2.x, s3.x);
      acc1(v[k].y, s1.y, s2.y, s3.y);
      acc1(v[k].z, s1.z, s2.z, s3.z);
      acc1(v[k].w, s1.w, s2.w, s3.w);
    }
  }
#endif

  // Reduce the 4 row-subgroups per column, then one plain store (deterministic)
  __shared__ float4 red[3][256];
  red[0][t] = s1;
  red[1][t] = s2;
  red[2][t] = s3;
  __syncthreads();
  if (g == 0) {
    float* dst = partial + (size_t)blockIdx.y * 3072;
#pragma unroll
    for (int comp = 0; comp < 3; ++comp) {
      float4 a = red[comp][t];
      float4 b = red[comp][t + 64];
      float4 c2 = red[comp][t + 128];
      float4 d2 = red[comp][t + 192];
      float4 s;
      s.x = (a.x + b.x) + (c2.x + d2.x);
      s.y = (a.y + b.y) + (c2.y + d2.y);
      s.z = (a.z + b.z) + (c2.z + d2.z);
      s.w = (a.w + b.w) + (c2.w + d2.w);
      *(float4*)(dst + comp * 1024 + colbase) = s;
    }
  }
}

// Single block, 1024 threads: thread j owns column j. Reduces nchunk partials
// in f64, applies inv = 1/max(sqrt(sum x^2),1e-12), contribution
// inv*(sum_xlogx + log(inv)*sum_x), tree-reduces 1024 doubles, writes scalar.
__global__ __launch_bounds__(1024) void entropy_finalize_kernel(
    const float* __restrict__ partial, int nchunk, float* __restrict__ out) {
  const int j = threadIdx.x;
  double a1 = 0.0, a2 = 0.0, a3 = 0.0;
  for (int y = 0; y < nchunk; ++y) {
    const float* p = partial + (size_t)y * 3072;
    a1 += (double)p[j];
    a2 += (double)p[1024 + j];
    a3 += (double)p[2048 + j];
  }
  const double nrm = sqrt(a2);
  const double inv = 1.0 / fmax(nrm, 1e-12);
  const double lg = log(inv);
  __shared__ double red[1024];
  red[j] = inv * (a3 + lg * a1);
  __syncthreads();
  for (int off = 512; off > 0; off >>= 1) {
    if (j < off) red[j] += red[j + off];
    __syncthreads();
  }
  if (j == 0) {
    const double rows = (double)nchunk * (double)ROWS_PER_CHUNK;
    const double S = red[0];
    const double epsCorr = rows * 1024.0 * 1e-8; // sum_{ij} p*log(1+eps/p) ~= N*eps
    out[0] = (float)(-(S + epsCorr) / rows);
  }
}

extern "C" void kernel_launch(void* const* d_in, const int* in_sizes, int n_in,
                              void* d_out, int out_size, void* d_ws,
                              size_t ws_size, hipStream_t stream) {
  (void)n_in;
  (void)out_size;
  (void)ws_size;
  const float* x = (const float*)d_in[0];
  float* out = (float*)d_out;
  float* ws = (float*)d_ws; // partial[nchunk][3][1024] floats (~1.5 MB)

  const long long total = (long long)in_sizes[0]; // rows * 1024
  const int rows = (int)(total / COLS);           // 65536
  const int nchunk = rows / ROWS_PER_CHUNK;       // 128

  dim3 grid(4, nchunk, 1);
  colstats_kernel<<<grid, 256, 0, stream>>>(x, ws);
  entropy_finalize_kernel<<<1, 1024, 0, stream>>>(ws, nchunk, out);
}